// BasicModel_37031208026271
// MI455X (gfx1250) — compile-verified
//
#include <hip/hip_runtime.h>
#include <stdint.h>

// Problem constants (from reference)
#define NN 100000      // NUM_NODES
#define NC 64          // NUM_CLASS + 1
#define NE 1600000     // NUM_EDGES

// d_out layout: [ poss_node : NN*NC floats ][ poss_edge copy : NE*NC floats ]

// ---------------------------------------------------------------------------
// Kernel 1: zero the accumulator region (harness poisons d_out; atomics need 0)
// Default RT stores -> lines land dirty in L2, ready for the atomic phase.
// ---------------------------------------------------------------------------
__global__ __launch_bounds__(256) void zero_kernel(float4* __restrict__ out) {
    const int i = blockIdx.x * 256 + threadIdx.x;
    out[i] = make_float4(0.f, 0.f, 0.f, 0.f);
}

// ---------------------------------------------------------------------------
// Kernel 2: fused stream + scatter.
// One wave32 handles 2 edges (2 x 64 floats = 512B = 32 lanes x 16B):
//   - async DMA global->LDS (NT: 410MB single-use stream, keep out of L2)
//   - metadata loads issued under the async DMA's shadow
//   - ds_load_b128 per lane for compute
//   - async DMA LDS->global copy-out (NT store: written once, never re-read)
//   - 4x global_atomic_add_f32 per lane into acc[src*64+c] (RT, L2-resident)
// 100000 blocks x 256 threads = 800000 waves = 1.6M edges exactly.
// ---------------------------------------------------------------------------
__global__ __launch_bounds__(256) void edge_kernel(
    const int*   __restrict__ edges,      // (NE,2) int32, row-major
    const float* __restrict__ weights,    // (NE,)
    const float* __restrict__ poss_edge,  // (NE,64)
    float*       __restrict__ acc,        // d_out[0 .. NN*64)
    float*       __restrict__ copy_out)   // d_out[NN*64 .. +NE*64)
{
    __shared__ float lds[8 * 128];                 // 8 waves * 512B = 4KB

    const int tid       = threadIdx.x;
    const int lane      = tid & 31;
    const int waveInBlk = tid >> 5;
    const long long pair = (long long)blockIdx.x * 8 + waveInBlk;  // edge pair id

    // This lane's 16B chunk of the 512B pair tile
    const long long goff = pair * 128 + (long long)lane * 4;       // float offset
    const unsigned long long gsrc =
        (unsigned long long)(uintptr_t)(poss_edge + goff);
    const unsigned long long gdst =
        (unsigned long long)(uintptr_t)(copy_out + goff);

    float* ldsp = &lds[waveInBlk * 128 + lane * 4];
    // Flat->LDS offset: low 32 bits of the flat address are the LDS address
    // (ISA 10.2 aperture mapping), matching async-op VDST semantics.
    const unsigned ldsa = (unsigned)(uintptr_t)ldsp;

    // Async DMA: global -> LDS, 16B per lane; non-temporal (single-use stream)
    asm volatile("global_load_async_to_lds_b128 %0, %1, off th:TH_LOAD_NT"
                 :: "v"(ldsa), "v"(gsrc) : "memory");

    // Metadata loads overlap the async DMA (kept after the asm by the
    // memory clobber, waited on only at first use in the atomics below).
    const int e   = (int)(pair * 2) + (lane >> 4);  // lanes 0-15: edge0, 16-31: edge1
    const float w = weights[e];
    const int src = edges[2 * e];                   // edges[e,0]

    asm volatile("s_wait_asynccnt 0" ::: "memory");

    // LDS -> VGPR for the scaled scatter (ds_load_b128)
    const float4 v = *(const float4*)ldsp;

    // Async DMA: LDS -> global copy-out; non-temporal store (never re-read).
    // Safe: the load above completed; S_ENDPGM's implicit wait-idle drains it.
    asm volatile("global_store_async_from_lds_b128 %0, %1, off th:TH_STORE_NT"
                 :: "v"(gdst), "v"(ldsa) : "memory");

    float* a = acc + (long long)src * NC + (lane & 15) * 4;
    unsafeAtomicAdd(a + 0, v.x * w);               // global_atomic_add_f32 (L2)
    unsafeAtomicAdd(a + 1, v.y * w);
    unsafeAtomicAdd(a + 2, v.z * w);
    unsafeAtomicAdd(a + 3, v.w * w);
}

// ---------------------------------------------------------------------------
// Kernel 3: poss_node /= neighbours_sum (broadcast over 64 classes).
// 1.6M float4; 16 consecutive float4 share one node -> broadcast read.
// IEEE division to match the reference's rounding.
// ---------------------------------------------------------------------------
__global__ __launch_bounds__(256) void norm_kernel(
    float4* __restrict__ poss_node, const float* __restrict__ nsum)
{
    const int i    = blockIdx.x * 256 + threadIdx.x;
    const int node = i >> 4;                       // 64 floats = 16 float4 per node
    const float s  = nsum[node];
    float4 v = poss_node[i];
    v.x /= s; v.y /= s; v.z /= s; v.w /= s;
    poss_node[i] = v;
}

// ---------------------------------------------------------------------------
extern "C" void kernel_launch(void* const* d_in, const int* in_sizes, int n_in,
                              void* d_out, int out_size, void* d_ws, size_t ws_size,
                              hipStream_t stream) {
    const int*   edges     = (const int*)d_in[0];
    const float* weights   = (const float*)d_in[1];
    const float* poss_edge = (const float*)d_in[2];
    const float* nsum      = (const float*)d_in[3];

    float* out       = (float*)d_out;
    float* poss_node = out;                         // NN*NC floats
    float* copy_out  = out + (size_t)NN * NC;       // NE*NC floats

    zero_kernel<<<(NN * NC) / (4 * 256), 256, 0, stream>>>((float4*)poss_node);
    edge_kernel<<<NE / 16, 256, 0, stream>>>(edges, weights, poss_edge,
                                             poss_node, copy_out);
    norm_kernel<<<(NN * NC) / (4 * 256), 256, 0, stream>>>((float4*)poss_node,
                                                           nsum);
}